// LaplacianPhysicsLoss_68607807586434
// MI455X (gfx1250) — compile-verified
//
#include <hip/hip_runtime.h>
#include <math.h>

#define MAXN 32
#define PHYS_MARGIN 0.1f

typedef float v2f __attribute__((ext_vector_type(2)));
typedef float v8f __attribute__((ext_vector_type(8)));

__device__ __forceinline__ float waveReduceSum(float v) {
#pragma unroll
    for (int off = 16; off > 0; off >>= 1) v += __shfl_xor(v, off, 32);
    return v;
}
__device__ __forceinline__ float waveReduceMin(float v) {
#pragma unroll
    for (int off = 16; off > 0; off >>= 1) { float o = __shfl_xor(v, off, 32); v = fminf(v, o); }
    return v;
}
__device__ __forceinline__ float waveReduceMax(float v) {
#pragma unroll
    for (int off = 16; off > 0; off >>= 1) { float o = __shfl_xor(v, off, 32); v = fmaxf(v, o); }
    return v;
}

__global__ __launch_bounds__(32)
void lap_init(float* ws) {
    if (threadIdx.x < 4) ws[threadIdx.x] = 0.0f;
}

// One wave (32 lanes) per 32x32 matrix. Lane i owns row i / column i as noted.
__global__ __launch_bounds__(32)
void lap_main(const float* __restrict__ predA,
              const float* __restrict__ targetA,
              const float* __restrict__ nodeMask,
              float* __restrict__ out,
              float* __restrict__ ws)
{
    __shared__ float As[MAXN][MAXN + 1];   // padded: stride 33 -> conflict-free rows & cols
    __shared__ float vbuf[MAXN];
    __shared__ float wbuf[MAXN];
    __shared__ float deg[MAXN];
    __shared__ float msk[MAXN];
    __shared__ float dd[MAXN];
    __shared__ float es[MAXN];

    const int b    = blockIdx.x;
    const int lane = threadIdx.x;
    const float* __restrict__ Ab = predA   + (size_t)b * (MAXN * MAXN);
    const float* __restrict__ Tb = targetA + (size_t)b * (MAXN * MAXN);

    const float m_l = nodeMask[(size_t)b * MAXN + lane];
    msk[lane] = m_l;
    __syncthreads();

    // ---- Stage pred_A into LDS (coalesced: lane = column, iterate rows) + fused BCE ----
    float bce_acc = 0.0f;
#pragma unroll 4
    for (int r = 0; r < MAXN; ++r) {
        float p = Ab[r * MAXN + lane];
        float t = Tb[r * MAXN + lane];
        As[r][lane] = p;
        float m2 = msk[r] * m_l;
        bce_acc += m2 * (t * __logf(p) + (1.0f - t) * __logf(1.0f - p));
    }
    float bce_sum = waveReduceSum(bce_acc);
    float sm      = waveReduceSum(m_l);
    if (lane == 0) {
        atomicAdd(&ws[0], -bce_sum);   // sum of masked BCE
        atomicAdd(&ws[1], sm * sm);    // sum of mask_2d
    }
    __syncthreads();

    // ---- Row sums (degrees) via V_WMMA_F32_16X16X4_F32: D = A * ones ----
    // A-fragment (16x4 f32): lanes 0-15 rows M=0..15 hold K={k0,k0+1}; lanes 16-31 hold K={k0+2,k0+3}.
    // B = all-ones -> C[m][n] = sum_k A[m][k], independent of B layout.
    {
        const int m  = lane & 15;
        const int kk = (lane >> 4) << 1;
        v2f bones; bones.x = 1.0f; bones.y = 1.0f;
#pragma unroll
        for (int bi = 0; bi < MAXN; bi += 16) {
            v8f acc = {0.0f, 0.0f, 0.0f, 0.0f, 0.0f, 0.0f, 0.0f, 0.0f};
#pragma unroll
            for (int k0 = 0; k0 < MAXN; k0 += 4) {
                v2f a;
                a.x = As[bi + m][k0 + kk];
                a.y = As[bi + m][k0 + kk + 1];
                acc = __builtin_amdgcn_wmma_f32_16x16x4_f32(
                        false, a, false, bones, (short)0, acc, false, false);
            }
            // C/D layout: VGPR r, lanes 0-15 -> (M=r, N=lane); lanes 16-31 -> (M=8+r, N=lane-16).
            // Column N=0 lives in lanes 0 and 16.
            if (lane == 0) {
#pragma unroll
                for (int r = 0; r < 8; ++r) deg[bi + r] = acc[r];
            }
            if (lane == 16) {
#pragma unroll
                for (int r = 0; r < 8; ++r) deg[bi + 8 + r] = acc[r];
            }
        }
    }
    __syncthreads();

    // ---- Build Laplacian in place: L = -A, L[i][i] += deg[i] + jitter[i] ----
    {
        const float jit0  = 1e-5f;
        const float jstep = (1e-4f - 1e-5f) / 31.0f;
#pragma unroll 4
        for (int r = 0; r < MAXN; ++r) {
            float a   = As[r][lane];
            float val = -a;
            if (r == lane) val += deg[r] + (jit0 + jstep * (float)r);
            As[r][lane] = val;
        }
    }
    __syncthreads();

    // ---- Householder tridiagonalization (lane i owns row i) ----
    for (int k = 0; k <= MAXN - 3; ++k) {
        const int i = lane;
        float xi    = (i >= k + 1) ? As[i][k] : 0.0f;
        float norm2 = waveReduceSum(xi * xi);      // uniform across wave
        float x0    = __shfl(xi, k + 1, 32);
        if (norm2 > 1e-28f) {                       // uniform branch
            float nrm   = sqrtf(norm2);
            float alpha = (x0 >= 0.0f) ? -nrm : nrm;
            float vi    = xi - ((i == k + 1) ? alpha : 0.0f);
            float vtv   = 2.0f * (norm2 - alpha * x0);
            float tau   = 2.0f / vtv;
            vbuf[i] = vi;
            __syncthreads();
            // p = tau * A v  (v supported on k+1..31)
            float pi = 0.0f;
            for (int j = k + 1; j < MAXN; ++j) pi += As[i][j] * vbuf[j];
            pi *= tau;
            float pv = waveReduceSum(pi * vi);
            float K  = 0.5f * tau * pv;
            float wi = pi - K * vi;
            if (i < k) wi = 0.0f;                  // keep finished tridiagonal rows exact
            wbuf[i] = wi;
            __syncthreads();
            // A <- A - v w^T - w v^T (only columns >= k can change)
            for (int j = k; j < MAXN; ++j)
                As[i][j] -= vi * wbuf[j] + wi * vbuf[j];
        }
        __syncthreads();
    }

    dd[lane] = As[lane][lane];
    if (lane < MAXN - 1) es[lane] = As[lane + 1][lane];
    __syncthreads();

    // ---- Gershgorin bounds ----
    float ePrev = (lane > 0)        ? es[lane - 1] : 0.0f;
    float eNext = (lane < MAXN - 1) ? es[lane]     : 0.0f;
    float rad   = fabsf(ePrev) + fabsf(eNext);
    float lo = waveReduceMin(dd[lane] - rad);
    float hi = waveReduceMax(dd[lane] + rad);

    // ---- 33-way multisection with Sturm counts: find lambda_2 (2nd smallest) ----
#pragma unroll 1
    for (int iter = 0; iter < 8; ++iter) {
        float step = (hi - lo) * (1.0f / 33.0f);
        float x    = lo + step * (float)(lane + 1);
        float q    = dd[0] - x;
        int   cnt  = (q < 0.0f) ? 1 : 0;
#pragma unroll 1
        for (int t = 1; t < MAXN; ++t) {
            float e  = es[t - 1];
            float qs = (fabsf(q) < 1e-30f) ? ((q < 0.0f) ? -1e-30f : 1e-30f) : q;
            q = (dd[t] - x) - (e * e) / qs;
            cnt += (q < 0.0f) ? 1 : 0;
        }
        unsigned int b32 = (unsigned int)__ballot(cnt >= 2);  // wave32: low 32 bits
        if (b32 == 0u) {
            lo = lo + step * 32.0f;
        } else {
            int s = __ffs(b32) - 1;          // first lane with count >= 2
            hi = lo + step * (float)(s + 1);
            lo = lo + step * (float)s;       // previous probe (or unchanged lo if s==0)
        }
    }
    float lambda2 = 0.5f * (lo + hi);

    if (lane == 0) {
        out[2 + b] = lambda2;
        atomicAdd(&ws[2], fmaxf(PHYS_MARGIN - lambda2, 0.0f));
    }
}

__global__ __launch_bounds__(32)
void lap_finalize(const float* __restrict__ ws, float* __restrict__ out, int batch) {
    if (threadIdx.x == 0 && blockIdx.x == 0) {
        out[0] = ws[0] / fmaxf(ws[1], 1.0f);          // bce_loss
        out[1] = ws[2] / (float)batch;                // physics_loss
    }
}

extern "C" void kernel_launch(void* const* d_in, const int* in_sizes, int n_in,
                              void* d_out, int out_size, void* d_ws, size_t ws_size,
                              hipStream_t stream)
{
    const float* predA    = (const float*)d_in[0];
    const float* targetA  = (const float*)d_in[1];
    const float* nodeMask = (const float*)d_in[2];
    float* out = (float*)d_out;
    float* ws  = (float*)d_ws;

    const int batch = in_sizes[0] / (MAXN * MAXN);

    lap_init<<<1, 32, 0, stream>>>(ws);
    lap_main<<<batch, 32, 0, stream>>>(predA, targetA, nodeMask, out, ws);
    lap_finalize<<<1, 32, 0, stream>>>(ws, out, batch);
}